// DataTermLayer_61546881351712
// MI455X (gfx1250) — compile-verified
//
#include <hip/hip_runtime.h>
#include <hip/hip_bf16.h>

// Optical-flow data-term update (DataTermLayer), MI455X / gfx1250.
// Bandwidth-bound gather+stream kernel: ~384 MB min traffic, ~0.5 GFLOP
// => HBM-roofline at 23.3 TB/s (~16.5 us). No matrix contraction -> WMMA N/A.
// gfx1250 strategy: wave32, 1 thread = 4 consecutive pixels, 128-bit global
// loads/stores, non-temporal TH hints on stream-once arrays (I2,u,v,outputs)
// so I1 (the only reused array: gradients + 4 bilinear corners) stays hot in
// WGP$/L2. Index math kept unsigned so pow2 div/mod lower to single AND/SHR.

typedef __attribute__((ext_vector_type(4))) float v4f;

#define ALPHA 0.15f

template <unsigned LOGW>
__device__ __forceinline__ float bilerp_fetch(const float* __restrict__ img,
                                              float yf, float xf,
                                              float Hm1, float Wm1) {
    // Matches reference gather(): zero if corner outside [0,W-1]x[0,H-1],
    // else clamp + load. xf/yf are integer-valued floats (floor results).
    bool valid = (xf >= 0.0f) && (xf <= Wm1) && (yf >= 0.0f) && (yf <= Hm1);
    unsigned xi = (unsigned)fminf(fmaxf(xf, 0.0f), Wm1);
    unsigned yi = (unsigned)fminf(fmaxf(yf, 0.0f), Hm1);
    float val = img[(yi << LOGW) + xi];
    return valid ? val : 0.0f;
}

template <unsigned LOGH, unsigned LOGW>
__global__ __launch_bounds__(256) void dataterm_kernel(
    const float* __restrict__ I1, const float* __restrict__ I2,
    const float* __restrict__ U,  const float* __restrict__ V,
    float* __restrict__ Unext, float* __restrict__ Vnext, unsigned N) {

    constexpr unsigned H  = 1u << LOGH;
    constexpr unsigned W  = 1u << LOGW;
    constexpr unsigned HW = H * W;

    const unsigned tid  = blockIdx.x * blockDim.x + threadIdx.x;
    const unsigned base = tid << 2;                 // 4 pixels per thread
    if (base >= N) return;

    // Pure mask/shift decomposition (unsigned => no sign-fixup sequences).
    const unsigned w     = base & (W - 1u);
    const unsigned h     = (base >> LOGW) & (H - 1u);
    const unsigned bbase = base & ~(HW - 1u);       // b * H * W

    const float* __restrict__ I1b = I1 + bbase;

    // I1 tile (cached/reused): center float4, guarded right scalar, row-below float4.
    v4f i1c = *(const v4f*)(I1 + base);
    float i1r = 0.0f;
    const bool has_right = (w + 4u < W);            // lane-3 horizontal neighbor
    if (has_right) i1r = I1[base + 4u];
    v4f i1d = (v4f)0.0f;
    const bool has_down = (h < H - 1u);
    if (has_down) i1d = *(const v4f*)(I1 + base + W);

    // Stream-once arrays: non-temporal 128-bit loads (th:TH_NT).
    v4f uu = __builtin_nontemporal_load((const v4f*)(U  + base));
    v4f vv = __builtin_nontemporal_load((const v4f*)(V  + base));
    v4f i2 = __builtin_nontemporal_load((const v4f*)(I2 + base));

    const float Hm1 = (float)(H - 1u);
    const float Wm1 = (float)(W - 1u);
    const float hf  = (float)h;

    float i1a[5] = { i1c.x, i1c.y, i1c.z, i1c.w, i1r };
    float dna[4] = { i1d.x, i1d.y, i1d.z, i1d.w };
    float ua[4]  = { uu.x, uu.y, uu.z, uu.w };
    float va[4]  = { vv.x, vv.y, vv.z, vv.w };
    float i2a[4] = { i2.x, i2.y, i2.z, i2.w };

    float una[4], vna[4];

#pragma unroll
    for (int k = 0; k < 4; ++k) {
        // NOTE reference swap: grad_x = dy (vertical), grad_y = dx (horizontal).
        const float gx = has_down ? (dna[k] - i1a[k]) : 0.0f;
        // Lanes 0..2 are always interior (base % 4 == 0, W % 4 == 0);
        // only lane 3 can touch the last column.
        const float gy = (k < 3) ? (i1a[k + 1] - i1a[k])
                                 : (has_right ? (i1r - i1a[3]) : 0.0f);

        // Bilinear warp of I1 at (w+k + 0.5u, h + 0.5v), zero padding.
        const float x  = (float)(w + (unsigned)k) + 0.5f * ua[k];
        const float y  = hf + 0.5f * va[k];
        const float x0 = floorf(x);
        const float y0 = floorf(y);
        const float x1 = x0 + 1.0f;
        const float y1 = y0 + 1.0f;
        const float wx1 = x - x0, wx0 = 1.0f - wx1;
        const float wy1 = y - y0, wy0 = 1.0f - wy1;

        const float f00 = bilerp_fetch<LOGW>(I1b, y0, x0, Hm1, Wm1);
        const float f01 = bilerp_fetch<LOGW>(I1b, y0, x1, Hm1, Wm1);
        const float f10 = bilerp_fetch<LOGW>(I1b, y1, x0, Hm1, Wm1);
        const float f11 = bilerp_fetch<LOGW>(I1b, y1, x1, Hm1, Wm1);

        const float warped = f00 * (wy0 * wx0) + f01 * (wy0 * wx1)
                           + f10 * (wy1 * wx0) + f11 * (wy1 * wx1);
        const float dT = warped - i2a[k];

        una[k] = ua[k] - ALPHA * dT * gx;
        vna[k] = va[k] - ALPHA * dT * gy;
    }

    v4f uo = { una[0], una[1], una[2], una[3] };
    v4f vo = { vna[0], vna[1], vna[2], vna[3] };
    // Non-temporal 128-bit stores: outputs are write-once, keep L2 for I1.
    __builtin_nontemporal_store(uo, (v4f*)(Unext + base));
    __builtin_nontemporal_store(vo, (v4f*)(Vnext + base));
}

extern "C" void kernel_launch(void* const* d_in, const int* in_sizes, int n_in,
                              void* d_out, int out_size, void* d_ws, size_t ws_size,
                              hipStream_t stream) {
    const float* I1 = (const float*)d_in[0];
    const float* I2 = (const float*)d_in[1];
    const float* U  = (const float*)d_in[2];
    const float* V  = (const float*)d_in[3];

    const unsigned N = (unsigned)in_sizes[0];   // B*H*W*C = 16*1024*1024
    float* out   = (float*)d_out;
    float* Unext = out;                          // tuple order: (u_next, v_next)
    float* Vnext = out + N;

    const unsigned threads = N / 4u;             // 4 pixels per thread
    const unsigned block   = 256u;               // 8 wave32 per block
    const unsigned grid    = (threads + block - 1u) / block;

    hipLaunchKernelGGL((dataterm_kernel<10, 10>), dim3(grid), dim3(block), 0, stream,
                       I1, I2, U, V, Unext, Vnext, N);
}